// CenterLoss_label_33277406609679
// MI455X (gfx1250) — compile-verified
//
#include <hip/hip_runtime.h>

typedef __attribute__((ext_vector_type(2))) float v2f;
typedef __attribute__((ext_vector_type(4))) float v4f;
typedef __attribute__((ext_vector_type(8))) float v8f;

#define KSPLIT 4  // waves splitting the D dimension per 16-row tile

__global__ void center_loss_zero(float* out) { out[0] = 0.0f; }

__global__ __launch_bounds__(256) void center_loss_wmma(
    const int* __restrict__ y,
    const float* __restrict__ hidden,
    const float* __restrict__ centers,
    float* __restrict__ out,
    int D, int niter, float scale)
{
  const int lane = threadIdx.x & 31;
  const int wave = blockIdx.x * (blockDim.x >> 5) + (threadIdx.x >> 5);

  const int tile    = wave / KSPLIT;     // 16-row tile
  const int kslice  = wave % KSPLIT;     // K-range of the tile
  const int rowBase = tile * 16;

  // f32 16x4 A-fragment: lane L -> row M = L%16; VGPR j -> K = 2*(L>=16)+j.
  // Column->K-slot assignment is free for our purposes (diagonal is a
  // permutation-invariant sum of squares), so each lane streams a float4 at
  // k0 + 4*(lane>=16): d.xy feeds WMMA #1, d.zw feeds WMMA #2.
  const int m    = lane & 15;
  const int row  = rowBase + m;
  const int half = lane >> 4;            // 0: lanes 0-15, 1: lanes 16-31

  const int label = y[row];
  const int kbeg  = kslice * (D / KSPLIT);

  const float* __restrict__ hp = hidden  + (size_t)row   * D + kbeg + 4 * half;
  const float* __restrict__ cp = centers + (size_t)label * D + kbeg + 4 * half;

  // Prefetch the gathered center row (global_prefetch_b8).
  __builtin_prefetch(cp, 0, 1);

  v8f acc = {};
  #pragma unroll 4
  for (int i = 0; i < niter; ++i) {       // niter is wave-uniform -> scalar loop
    v4f h = *(const v4f*)hp;
    v4f c = *(const v4f*)cp;
    hp += 8;                              // 8 columns consumed per iteration
    cp += 8;
    v4f d = h - c;
    v2f dlo = __builtin_shufflevector(d, d, 0, 1);
    v2f dhi = __builtin_shufflevector(d, d, 2, 3);
    acc = __builtin_amdgcn_wmma_f32_16x16x4_f32(
        false, dlo, false, dlo, (short)0, acc, false, false);
    acc = __builtin_amdgcn_wmma_f32_16x16x4_f32(
        false, dhi, false, dhi, (short)0, acc, false, false);
  }

  // trace(acc): D[v][v] at lane v (v<8), D[v+8][v+8] at lane v+24, VGPR v.
  float t = 0.0f;
  #pragma unroll
  for (int v = 0; v < 8; ++v) {
    t += __shfl(acc[v], v,      32);
    t += __shfl(acc[v], v + 24, 32);
  }

  if (lane == 0) {
    atomicAdd(out, t * scale);            // global_atomic_add_f32
  }
}

extern "C" void kernel_launch(void* const* d_in, const int* in_sizes, int n_in,
                              void* d_out, int out_size, void* d_ws, size_t ws_size,
                              hipStream_t stream) {
  const int*   y       = (const int*)d_in[0];
  const float* hidden  = (const float*)d_in[1];
  const float* centers = (const float*)d_in[2];
  float*       out     = (float*)d_out;

  const int B = in_sizes[0];             // 1024
  const int D = in_sizes[1] / B;         // 512

  center_loss_zero<<<1, 1, 0, stream>>>(out);

  const int   niter = (D / KSPLIT) / 8;  // 16 iterations of 8 columns
  const float scale = 0.5f / (float)B;

  const int waves   = (B / 16) * KSPLIT; // 256 waves
  const int threads = 256;               // 8 waves per block
  const int blocks  = (waves * 32 + threads - 1) / threads;  // 32 blocks
  center_loss_wmma<<<blocks, threads, 0, stream>>>(y, hidden, centers, out,
                                                   D, niter, scale);
}